// SoftMaskingModule_60816736911760
// MI455X (gfx1250) — compile-verified
//
#include <hip/hip_runtime.h>
#include <hip/hip_bf16.h>
#include <stdint.h>

// Problem constants (match reference)
#define BD 2
#define SD 2048
#define VD 32000
#define HD 1024
#define MASKID 31999
#define NT 256                 // threads per block (8 wave32)
#define CHUNK 1024             // floats per async chunk = NT * 4
#define NCHUNKS 31             // 31 * 1024 = 31744; MASKID lives in the tail
#define TAILBASE 31744         // tail: 256 floats -> indices 31744..31999
#define NBUF 4                 // async pipeline depth (4 x 4KB LDS)

// ---------- CDNA5 async global->LDS helpers (gfx1250) ----------
__device__ __forceinline__ void async_ld_b128(uint32_t lds_addr, uint32_t voff,
                                              const float* sbase) {
  // GVS mode: global addr = SGPR64 base + VGPR32 offset; VDST = LDS byte addr
  asm volatile("global_load_async_to_lds_b128 %0, %1, %2"
               :: "v"(lds_addr), "v"(voff), "s"(sbase)
               : "memory");
}
__device__ __forceinline__ void wait_async0() { asm volatile("s_wait_asynccnt 0x0" ::: "memory"); }
__device__ __forceinline__ void wait_async1() { asm volatile("s_wait_asynccnt 0x1" ::: "memory"); }
__device__ __forceinline__ void wait_async2() { asm volatile("s_wait_asynccnt 0x2" ::: "memory"); }
__device__ __forceinline__ void wait_async3() { asm volatile("s_wait_asynccnt 0x3" ::: "memory"); }
__device__ __forceinline__ void wait_ds0()    { asm volatile("s_wait_dscnt 0x0"    ::: "memory"); }

// merge ordering: larger value wins; ties -> lower index (matches lax.top_k)
__device__ __forceinline__ bool topgt(float av, int ai, float bv, int bi) {
  return (av > bv) || ((av == bv) && (ai < bi));
}

__device__ __forceinline__ float softplus_f(float x) {
  return (x > 20.f) ? x : log1pf(expf(x));
}

__global__ __launch_bounds__(NT) void softmask_kernel(
    const int* __restrict__ x_t,
    const float* __restrict__ probs,
    const float* __restrict__ emb,
    const float* __restrict__ omega_s,
    const float* __restrict__ omega_a,
    const float* __restrict__ omega_b,
    float* __restrict__ out) {
  __shared__ __align__(16) float pbuf[NBUF][CHUNK];  // 16 KB ring of DMA buffers
  __shared__ float cv[NT * 3];
  __shared__ int   ci[NT * 3];
  __shared__ float ce[NT];
  __shared__ int   g_thr_i;     // float bits of shared lower bound on global 3rd-largest
  __shared__ int   r_idx[3];
  __shared__ float r_w[3];
  __shared__ float r_lam;

  const int t = threadIdx.x;
  const int token = blockIdx.x;                      // 0 .. B*S-1
  const float* pr = probs + (size_t)token * VD;

  const uint32_t lds_base = (uint32_t)(uintptr_t)(&pbuf[0][0]); // LDS offset = low 32 bits
  const uint32_t my16 = (uint32_t)t * 16u;

  // per-thread top-3 (sorted descending) + entropy accumulator
  float v0 = -1.f, v1 = -1.f, v2 = -1.f;
  int   i0 = 0x7fffffff, i1 = 0x7fffffff, i2 = 0x7fffffff;
  float ent = 0.f;

  // fully general branchless insert into sorted top-3 (pure v_cndmask)
  auto insert3 = [&](float p, int idx) {
    const bool gt2 = (p > v2);
    const bool gt1 = (p > v1);
    const bool gt0 = (p > v0);
    v2 = gt1 ? v1 : (gt2 ? p : v2);
    i2 = gt1 ? i1 : (gt2 ? idx : i2);
    v1 = gt0 ? v0 : (gt1 ? p : v1);
    i1 = gt0 ? i0 : (gt1 ? idx : i1);
    v0 = gt0 ? p : v0;
    i0 = gt0 ? idx : i0;
  };

  auto entropy_add = [&](float p) {
    const float lg = __logf(p);                      // trans pipe, co-executes
    ent = (p > 0.f) ? fmaf(-p, lg, ent) : ent;       // branchless guard
  };

  auto process_chunk = [&](int c) {
    const float4 p4 = *(const float4*)(&pbuf[c & (NBUF - 1)][t * 4]);
    // direct LDS access -> ds_load_b32 broadcast; re-read each iteration is
    // forced by the asm volatile "memory" clobbers in the loop (no volatile
    // generic-pointer cast, which would demote this to a FLAT access).
    const float thr = __int_as_float(g_thr_i);
    wait_ds0();    // LDS reads retired before this buffer is DMA-overwritten later
    entropy_add(p4.x);
    entropy_add(p4.y);
    entropy_add(p4.z);
    entropy_add(p4.w);
    // cheap filter: only proceed if some element can beat the shared threshold.
    // thr <= (global 3rd-largest - 1ulp) at all times, so nothing needed is lost.
    const float m4 = fmaxf(fmaxf(p4.x, p4.y), fmaxf(p4.z, p4.w));
    if (m4 > thr) {                                  // genuinely rare per wave
      const int base = c * CHUNK + t * 4;            // never contains MASKID
      insert3(p4.x, base + 0);
      insert3(p4.y, base + 1);
      insert3(p4.z, base + 2);
      insert3(p4.w, base + 3);
      // publish conservative bound: my 3rd-largest minus 1 ulp (int-bit trick:
      // positive-float bits order as signed ints; negative sentinels decode to
      // an even weaker threshold -> always safe).
      atomicMax(&g_thr_i, __float_as_int(v2) - 1);
    }
  };

  // init shared threshold, then start the DMA pipeline
  if (t == 0) g_thr_i = __float_as_int(-1.0f);
  async_ld_b128(lds_base + 0u * (CHUNK * 4u) + my16, 0u * (CHUNK * 4u) + my16, pr);
  async_ld_b128(lds_base + 1u * (CHUNK * 4u) + my16, 1u * (CHUNK * 4u) + my16, pr);
  async_ld_b128(lds_base + 2u * (CHUNK * 4u) + my16, 2u * (CHUNK * 4u) + my16, pr);
  __syncthreads();

  // steady state: issue chunk c+3, wait for chunk c (in-order completion), process
  for (int c = 0; c + 3 < NCHUNKS; ++c) {
    const uint32_t nb = (uint32_t)((c + 3) & (NBUF - 1));
    async_ld_b128(lds_base + nb * (CHUNK * 4u) + my16,
                  (uint32_t)(c + 3) * (CHUNK * 4u) + my16, pr);
    wait_async3();
    process_chunk(c);
  }
  // drain
  wait_async2(); process_chunk(NCHUNKS - 3);
  wait_async1(); process_chunk(NCHUNKS - 2);
  wait_async0(); process_chunk(NCHUNKS - 1);

  // tail: 256 remaining floats, direct coalesced loads (includes MASKID element)
  {
    const int idx = TAILBASE + t;
    const float p = pr[idx];
    entropy_add(p);
    if (idx != MASKID) insert3(p, idx);
  }

  // workgroup reduction: merge sorted 3-lists + entropy sum
  cv[t * 3 + 0] = v0; cv[t * 3 + 1] = v1; cv[t * 3 + 2] = v2;
  ci[t * 3 + 0] = i0; ci[t * 3 + 1] = i1; ci[t * 3 + 2] = i2;
  ce[t] = ent;
  __syncthreads();
  for (int s = NT / 2; s > 0; s >>= 1) {
    if (t < s) {
      const int o = t + s;
      float av[3] = {v0, v1, v2};
      int   ai[3] = {i0, i1, i2};
      float bv[3] = {cv[o * 3 + 0], cv[o * 3 + 1], cv[o * 3 + 2]};
      int   bi[3] = {ci[o * 3 + 0], ci[o * 3 + 1], ci[o * 3 + 2]};
      float rv[3]; int ri[3];
      int ia = 0, ib = 0;
#pragma unroll
      for (int k = 0; k < 3; ++k) {
        if (topgt(av[ia], ai[ia], bv[ib], bi[ib])) {
          rv[k] = av[ia]; ri[k] = ai[ia]; ++ia;
        } else {
          rv[k] = bv[ib]; ri[k] = bi[ib]; ++ib;
        }
      }
      v0 = rv[0]; v1 = rv[1]; v2 = rv[2];
      i0 = ri[0]; i1 = ri[1]; i2 = ri[2];
      ent += ce[o];
      cv[t * 3 + 0] = v0; cv[t * 3 + 1] = v1; cv[t * 3 + 2] = v2;
      ci[t * 3 + 0] = i0; ci[t * 3 + 1] = i1; ci[t * 3 + 2] = i2;
      ce[t] = ent;
    }
    __syncthreads();
  }

  if (t == 0) {
    const float ws = fminf(fmaxf(omega_s[0], 0.f), 1.f);
    const float wa = softplus_f(omega_a[0]);
    const float wb = -softplus_f(omega_b[0]);
    const float z = wa * (-ent - wb);
    const float lam = ws / (1.f + expf(-z));
    const float inv = 1.f / (v0 + v1 + v2 + 1e-10f);
    r_idx[0] = i0; r_idx[1] = i1; r_idx[2] = i2;
    r_w[0] = v0 * inv; r_w[1] = v1 * inv; r_w[2] = v2 * inv;
    r_lam = lam;
  }
  __syncthreads();

  // emit H=1024 floats: 256 threads x float4, gathered from emb (4KB-aligned rows)
  const int tok = x_t[token];
  float* o = out + (size_t)token * HD;
  const int h0 = t * 4;
  if (tok == MASKID) {
    const float lam = r_lam;
    const float om = 1.f - lam;
    const float w0 = r_w[0] * lam, w1 = r_w[1] * lam, w2 = r_w[2] * lam;
    const float4 e0 = *(const float4*)(emb + (size_t)r_idx[0] * HD + h0);
    const float4 e1 = *(const float4*)(emb + (size_t)r_idx[1] * HD + h0);
    const float4 e2 = *(const float4*)(emb + (size_t)r_idx[2] * HD + h0);
    const float4 vm = *(const float4*)(emb + (size_t)MASKID * HD + h0);
    float4 r;
    r.x = om * vm.x + w0 * e0.x + w1 * e1.x + w2 * e2.x;
    r.y = om * vm.y + w0 * e0.y + w1 * e1.y + w2 * e2.y;
    r.z = om * vm.z + w0 * e0.z + w1 * e1.z + w2 * e2.z;
    r.w = om * vm.w + w0 * e0.w + w1 * e1.w + w2 * e2.w;
    *(float4*)(o + h0) = r;
  } else {
    *(float4*)(o + h0) = *(const float4*)(emb + (size_t)tok * HD + h0);
  }
}

extern "C" void kernel_launch(void* const* d_in, const int* in_sizes, int n_in,
                              void* d_out, int out_size, void* d_ws, size_t ws_size,
                              hipStream_t stream) {
  (void)in_sizes; (void)n_in; (void)d_ws; (void)ws_size; (void)out_size;
  const int*   x_t     = (const int*)d_in[0];
  const float* probs   = (const float*)d_in[1];
  const float* emb     = (const float*)d_in[2];
  const float* omega_s = (const float*)d_in[3];
  const float* omega_a = (const float*)d_in[4];
  const float* omega_b = (const float*)d_in[5];
  float* out = (float*)d_out;

  dim3 grid(BD * SD);   // one workgroup per token
  dim3 block(NT);
  softmask_kernel<<<grid, block, 0, stream>>>(x_t, probs, emb,
                                              omega_s, omega_a, omega_b, out);
}